// LatticeMP_Block_4879082848674
// MI455X (gfx1250) — compile-verified
//
#include <hip/hip_runtime.h>
#include <hip/hip_bf16.h>

typedef __attribute__((ext_vector_type(2))) float v2f;
typedef __attribute__((ext_vector_type(8))) float v8f;

#define H 64
#define WP1_FLOATS (8 * 32 * 64)   /* msg_w1 packed: 8 n-tiles x 32 ksteps x 64 */
#define WP2_FLOATS (9 * 32 * 64)   /* msg_w2 packed (cols padded 129->144)      */
#define WP_TOTAL (WP1_FLOATS + WP2_FLOATS) /* 34816 floats = 8704 float4 */

__device__ __forceinline__ v8f wmma4(v2f a, v2f b, v8f c) {
  // D = A(16x4 f32) * B(4x16 f32) + C(16x16 f32)
  return __builtin_amdgcn_wmma_f32_16x16x4_f32(false, a, false, b, (short)0, c,
                                               false, false);
}

__device__ __forceinline__ v8f splat8(float s) {
  v8f r;
#pragma unroll
  for (int i = 0; i < 8; i++) r[i] = s;
  return r;
}

__device__ __forceinline__ v2f ld2(const float* p) { return *(const v2f*)p; }

__device__ __forceinline__ void atomicMaxFloat(float* addr, float val) {
  // int-max for non-negative, uint-min for negative; valid with -inf init.
  if (val >= 0.0f)
    atomicMax((int*)addr, __float_as_int(val));
  else
    atomicMin((unsigned int*)addr, __float_as_uint(val));
}

// Async global->LDS copy of one 16B chunk (CDNA5 ASYNCcnt path).
__device__ __forceinline__ void async_copy_b128(const float* g, float* lds) {
  unsigned laddr = (unsigned)(size_t)lds;  // low 32 bits of generic = LDS offset
  asm volatile("global_load_async_to_lds_b128 %0, %1, off" ::"v"(laddr),
               "v"(g)
               : "memory");
}
__device__ __forceinline__ void wait_async0() {
  asm volatile("s_wait_asynccnt 0x0" ::: "memory");
}

// ---------------------------------------------------------------- init
__global__ void init_kernel(float* nfoc1, unsigned int* nfoc2_bits, int* deg,
                            int nNodes) {
  int i = blockIdx.x * blockDim.x + threadIdx.x;
  int tot = nNodes * H;
  if (i < tot) {
    nfoc1[i] = 0.0f;
    nfoc2_bits[i] = 0xFF800000u;  // -inf
  }
  if (i < nNodes) deg[i] = 0;
}

// -------------------------------------------------------- weight pack
// Fragment-major repack: dst[(n*KK+kk)*64 + lane*2 + j] =
//   src[(4*kk + (lane>>4)*2 + j)*N + n*16 + (lane&15)]  (0 beyond N cols)
// so a B fragment is one contiguous, lane-indexed b64 load.
__global__ void pack_weights_kernel(const float* __restrict__ src,
                                    float* __restrict__ dst, int K, int N,
                                    int Np) {
  int p = blockIdx.x * blockDim.x + threadIdx.x;
  int KK = K >> 2, NT = Np >> 4;
  int tot = NT * KK * 64;
  if (p >= tot) return;
  int n = p / (KK * 64);
  int r = p % (KK * 64);
  int kk = r >> 6;
  int q = r & 63;
  int lane = q >> 1, j = q & 1;
  int k = 4 * kk + ((lane >> 4) << 1) + j;
  int col = n * 16 + (lane & 15);
  dst[p] = (col < N) ? src[k * N + col] : 0.0f;
}

// ------------------------------------------------- node pre: x and x1
// x = ((nf@rw1+rb1)@rw2+rb2)+nf ; x1 = x@l1w+l1b   (packed-fragment B)
__global__ void node_pre_kernel(const float* __restrict__ nf,
                                const float* __restrict__ prw1,
                                const float* __restrict__ rb1,
                                const float* __restrict__ prw2,
                                const float* __restrict__ rb2,
                                const float* __restrict__ pl1w,
                                const float* __restrict__ l1b,
                                float* __restrict__ ws_x,
                                float* __restrict__ ws_x1, int nNodes) {
  __shared__ float s_t[4][16][H];  // wave-private staging (16KB)
  int lane = threadIdx.x & 31;
  int w = threadIdx.x >> 5;
  int tile = blockIdx.x * 4 + w;
  if (tile * 16 >= nNodes) return;
  int row = lane & 15, hi = lane >> 4, hi2 = hi * 2;
  int node = tile * 16 + row;

  v2f a1[16];
#pragma unroll
  for (int kk = 0; kk < 16; kk++) a1[kk] = ld2(&nf[node * H + 4 * kk + hi2]);

  // GEMM1: t = nf @ rw1 + rb1 -> LDS
  for (int n = 0; n < 4; n++) {
    int col = n * 16 + row;
    v8f acc = splat8(rb1[col]);
#pragma unroll
    for (int kk = 0; kk < 16; kk++)
      acc = wmma4(a1[kk], ld2(&prw1[(n * 16 + kk) * 64 + lane * 2]), acc);
#pragma unroll
    for (int v = 0; v < 8; v++) s_t[w][v + 8 * hi][col] = acc[v];
  }

  // GEMM2: x = t @ rw2 + rb2 + nf
  v2f a2[16];
#pragma unroll
  for (int kk = 0; kk < 16; kk++) a2[kk] = ld2(&s_t[w][row][4 * kk + hi2]);
  for (int n = 0; n < 4; n++) {
    int col = n * 16 + row;
    v8f acc = splat8(rb2[col]);
#pragma unroll
    for (int kk = 0; kk < 16; kk++)
      acc = wmma4(a2[kk], ld2(&prw2[(n * 16 + kk) * 64 + lane * 2]), acc);
#pragma unroll
    for (int v = 0; v < 8; v++) {
      int m = tile * 16 + v + 8 * hi;
      float xv = acc[v] + nf[m * H + col];  // residual
      ws_x[m * H + col] = xv;
      s_t[w][v + 8 * hi][col] = xv;
    }
  }

  // GEMM3: x1 = x @ l1w + l1b
  v2f a3[16];
#pragma unroll
  for (int kk = 0; kk < 16; kk++) a3[kk] = ld2(&s_t[w][row][4 * kk + hi2]);
  for (int n = 0; n < 4; n++) {
    int col = n * 16 + row;
    v8f acc = splat8(l1b[col]);
#pragma unroll
    for (int kk = 0; kk < 16; kk++)
      acc = wmma4(a3[kk], ld2(&pl1w[(n * 16 + kk) * 64 + lane * 2]), acc);
#pragma unroll
    for (int v = 0; v < 8; v++) {
      int m = tile * 16 + v + 8 * hi;
      ws_x1[m * H + col] = acc[v];
    }
  }
}

// -------------------------------------------- edge MLP + scatter
// 256 threads = 8 waves, 16 edges/wave. Packed w1+w2 async-staged to LDS.
#define EW 8
__global__ void edge_kernel(const float* __restrict__ x,
                            const int* __restrict__ src,
                            const int* __restrict__ dst,
                            const float* __restrict__ wp,  // wp1 then wp2
                            const float* __restrict__ b1,
                            const float* __restrict__ b2,
                            float* __restrict__ nfoc1,
                            float* __restrict__ nfoc2, int* __restrict__ deg,
                            int nEdges) {
  extern __shared__ char smem_raw[];
  float* s_w1p = (float*)smem_raw;                // 16384 f (64KB)
  float* s_w2p = s_w1p + WP1_FLOATS;              // 18432 f (72KB)
  float* s_h = s_w2p + WP2_FLOATS;                // 8*16*128 f (64KB)
  int* s_dst = (int*)(s_h + EW * 16 * 128);       // 8*16
  float* s_k = (float*)(s_dst + EW * 16);         // 8*16

  int tid = threadIdx.x;
  // ---- async stage packed weights (8704 x 16B), overlaps feature gather
  for (int i = tid; i < WP_TOTAL / 4; i += 256)
    async_copy_b128(wp + (size_t)i * 4, s_w1p + (size_t)i * 4);

  int lane = tid & 31;
  int w = tid >> 5;
  int tile = blockIdx.x * EW + w;
  int ebase = tile * 16;
  int row = lane & 15, hi = lane >> 4, hi2 = hi * 2;
  int e = ebase + row;
  if (e >= nEdges) e = nEdges > 0 ? nEdges - 1 : 0;  // clamp (guarded later)
  int sN = src[e], dN = dst[e];
  if (lane < 16 && (ebase + lane) < nEdges) {
    s_dst[w * 16 + lane] = dN;
    atomicAdd(&deg[dN], 1);
  }

  // A fragments: concat(x[src], x[dst]) over K=128 (L2-resident gather)
  v2f a1[32];
#pragma unroll
  for (int kk = 0; kk < 32; kk++) {
    int k4 = 4 * kk;
    const float* base =
        (k4 < H) ? (x + (size_t)sN * H + k4) : (x + (size_t)dN * H + (k4 - H));
    a1[kk] = ld2(base + hi2);
  }

  wait_async0();
  __syncthreads();  // all waves: weights staged
  if (ebase >= nEdges) return;  // safe: no barriers after this point

  // layer 1: hmid = leaky_relu(e @ w1 + b1) -> LDS
  for (int n = 0; n < 8; n++) {
    int col = n * 16 + row;
    v8f acc = splat8(b1[col]);
#pragma unroll
    for (int kk = 0; kk < 32; kk++)
      acc = wmma4(a1[kk], ld2(&s_w1p[(n * 32 + kk) * 64 + lane * 2]), acc);
#pragma unroll
    for (int v = 0; v < 8; v++) {
      float t = acc[v];
      t = t > 0.0f ? t : 0.2f * t;  // leaky_relu(0.2)
      s_h[(w * 16 + v + 8 * hi) * 128 + col] = t;
    }
  }

  // layer 2 A fragments from LDS (transposed access)
  v2f a2[32];
#pragma unroll
  for (int kk = 0; kk < 32; kk++)
    a2[kk] = ld2(&s_h[(w * 16 + row) * 128 + 4 * kk + hi2]);

  // layer 2: o = hmid @ w2 + b2 over 129 cols (padded to 9x16 tiles)
  for (int n = 0; n < 9; n++) {
    int col = n * 16 + row;
    bool cv = col < 129;
    int colc = cv ? col : 128;
    float msk = cv ? 1.0f : 0.0f;
    v8f acc = splat8(b2[colc] * msk);
#pragma unroll
    for (int kk = 0; kk < 32; kk++)
      acc = wmma4(a2[kk], ld2(&s_w2p[(n * 32 + kk) * 64 + lane * 2]), acc);
    if (n == 0 && row == 0) {  // lanes with N==0 hold the gate column
#pragma unroll
      for (int v = 0; v < 8; v++)
        s_k[w * 16 + v + 8 * hi] = 1.0f / (1.0f + __expf(-acc[v]));
    }
    if (col >= 1 && col <= 128) {
#pragma unroll
      for (int v = 0; v < 8; v++) {
        int m = v + 8 * hi;
        if (ebase + m < nEdges) {
          float val = acc[v] * s_k[w * 16 + m];
          int d = s_dst[w * 16 + m];
          if (col <= 64)
            atomicAdd(&nfoc1[(size_t)d * H + (col - 1)], val);  // sum f1
          else
            atomicMaxFloat(&nfoc2[(size_t)d * H + (col - 65)], val);  // max f2
        }
      }
    }
  }
}

// ---------------------------------- node post: red + lin2 + final add
__global__ void node_post_kernel(
    const float* __restrict__ ws_x, const float* __restrict__ ws_x1,
    const float* __restrict__ nfoc1, const float* __restrict__ nfoc2,
    const int* __restrict__ deg, const float* __restrict__ predw,
    const float* __restrict__ redb, const float* __restrict__ pl2w,
    const float* __restrict__ l2b, float* __restrict__ out, int nNodes) {
  __shared__ float s_t[4][16][H];
  int lane = threadIdx.x & 31;
  int w = threadIdx.x >> 5;
  int tile = blockIdx.x * 4 + w;
  if (tile * 16 >= nNodes) return;
  int row = lane & 15, hi = lane >> 4, hi2 = hi * 2;
  int node = tile * 16 + row;

  // A over K=192: [x | nfoc1 | nfoc2(masked by deg)]
  v2f a[48];
#pragma unroll
  for (int kk = 0; kk < 48; kk++) {
    int k = 4 * kk + hi2;
    v2f v;
    if (k < 64)
      v = ld2(&ws_x[(size_t)node * H + k]);
    else if (k < 128)
      v = ld2(&nfoc1[(size_t)node * H + (k - 64)]);
    else
      v = ld2(&nfoc2[(size_t)node * H + (k - 128)]);
    a[kk] = v;
  }
  if (deg[node] <= 0) {  // isolated node: nfoc2 -> 0
#pragma unroll
    for (int kk = 32; kk < 48; kk++) {
      a[kk].x = 0.0f;
      a[kk].y = 0.0f;
    }
  }

  // new_x = concat @ red_w + red_b -> LDS
  for (int n = 0; n < 4; n++) {
    int col = n * 16 + row;
    v8f acc = splat8(redb[col]);
#pragma unroll
    for (int kk = 0; kk < 48; kk++)
      acc = wmma4(a[kk], ld2(&predw[(n * 48 + kk) * 64 + lane * 2]), acc);
#pragma unroll
    for (int v = 0; v < 8; v++) s_t[w][v + 8 * hi][col] = acc[v];
  }

  // x2 = new_x @ l2w + l2b ; out = x1 + x2
  v2f a2[16];
#pragma unroll
  for (int kk = 0; kk < 16; kk++) a2[kk] = ld2(&s_t[w][row][4 * kk + hi2]);
  for (int n = 0; n < 4; n++) {
    int col = n * 16 + row;
    v8f acc = splat8(l2b[col]);
#pragma unroll
    for (int kk = 0; kk < 16; kk++)
      acc = wmma4(a2[kk], ld2(&pl2w[(n * 16 + kk) * 64 + lane * 2]), acc);
#pragma unroll
    for (int v = 0; v < 8; v++) {
      int m = tile * 16 + v + 8 * hi;
      out[(size_t)m * H + col] = acc[v] + ws_x1[(size_t)m * H + col];
    }
  }
}

extern "C" void kernel_launch(void* const* d_in, const int* in_sizes, int n_in,
                              void* d_out, int out_size, void* d_ws,
                              size_t ws_size, hipStream_t stream) {
  const float* nf = (const float*)d_in[0];
  const int* src = (const int*)d_in[1];
  const int* dst = (const int*)d_in[2];
  const float* rw1 = (const float*)d_in[3];
  const float* rb1 = (const float*)d_in[4];
  const float* rw2 = (const float*)d_in[5];
  const float* rb2 = (const float*)d_in[6];
  const float* l1w = (const float*)d_in[7];
  const float* l1b = (const float*)d_in[8];
  const float* l2w = (const float*)d_in[9];
  const float* l2b = (const float*)d_in[10];
  const float* mw1 = (const float*)d_in[11];
  const float* mb1 = (const float*)d_in[12];
  const float* mw2 = (const float*)d_in[13];
  const float* mb2 = (const float*)d_in[14];
  const float* redw = (const float*)d_in[15];
  const float* redb = (const float*)d_in[16];
  float* out = (float*)d_out;

  int nNodes = in_sizes[0] / H;
  int nEdges = in_sizes[1];
  size_t nh = (size_t)nNodes * H;

  float* ws_x = (float*)d_ws;
  float* ws_x1 = ws_x + nh;
  float* nfoc1 = ws_x1 + nh;
  float* nfoc2 = nfoc1 + nh;
  int* deg = (int*)(nfoc2 + nh);
  float* wp1 = (float*)(deg + ((nNodes + 15) & ~15));  // wp1+wp2 contiguous
  float* wp2 = wp1 + WP1_FLOATS;
  float* p_rw1 = wp2 + WP2_FLOATS;
  float* p_rw2 = p_rw1 + 4 * 16 * 64;
  float* p_l1w = p_rw2 + 4 * 16 * 64;
  float* p_l2w = p_l1w + 4 * 16 * 64;
  float* p_redw = p_l2w + 4 * 16 * 64;

  int tot = nNodes * H;
  init_kernel<<<(tot + 255) / 256, 256, 0, stream>>>(
      nfoc1, (unsigned int*)nfoc2, deg, nNodes);

  // repack all weight matrices into fragment-major layout
  pack_weights_kernel<<<(WP1_FLOATS + 255) / 256, 256, 0, stream>>>(
      mw1, wp1, 128, 128, 128);
  pack_weights_kernel<<<(WP2_FLOATS + 255) / 256, 256, 0, stream>>>(
      mw2, wp2, 128, 129, 144);
  pack_weights_kernel<<<(4096 + 255) / 256, 256, 0, stream>>>(rw1, p_rw1, 64,
                                                              64, 64);
  pack_weights_kernel<<<(4096 + 255) / 256, 256, 0, stream>>>(rw2, p_rw2, 64,
                                                              64, 64);
  pack_weights_kernel<<<(4096 + 255) / 256, 256, 0, stream>>>(l1w, p_l1w, 64,
                                                              64, 64);
  pack_weights_kernel<<<(4096 + 255) / 256, 256, 0, stream>>>(l2w, p_l2w, 64,
                                                              64, 64);
  pack_weights_kernel<<<(12288 + 255) / 256, 256, 0, stream>>>(redw, p_redw,
                                                               192, 64, 64);

  int nodeTiles = (nNodes + 15) / 16;
  int nodeBlocks = (nodeTiles + 3) / 4;
  node_pre_kernel<<<nodeBlocks, 128, 0, stream>>>(
      nf, p_rw1, rb1, p_rw2, rb2, p_l1w, l1b, ws_x, ws_x1, nNodes);

  int edgeBlocks = (nEdges + 16 * EW - 1) / (16 * EW);
  size_t smem = (size_t)(WP_TOTAL + EW * 16 * 128) * 4 + EW * 16 * 8;
  edge_kernel<<<edgeBlocks, 32 * EW, smem, stream>>>(
      ws_x, src, dst, wp1, mb1, mb2, nfoc1, nfoc2, deg, nEdges);

  node_post_kernel<<<nodeBlocks, 128, 0, stream>>>(
      ws_x, ws_x1, nfoc1, nfoc2, deg, p_redw, redb, p_l2w, l2b, out, nNodes);
}